// CGCNN_85323820302956
// MI455X (gfx1250) — compile-verified
//
#include <hip/hip_runtime.h>
#include <hip/hip_bf16.h>

typedef float v2f __attribute__((ext_vector_type(2)));
typedef float v8f __attribute__((ext_vector_type(8)));

#define EPS 1e-5f
#define TM 64      // edge rows per block
#define CR 169     // 2E + Fe
#define CP 172     // padded K (multiple of 4)
#define FE 41
#define FV 92
#define EDIM 64
#define NG 256

#define PK_EDGE ((CP / 4) * 2 * 128)   // 11008 floats per packed edge matrix
#define PK_EMB  ((FV / 4) * 2 * 128)   // 5888 floats packed embedding matrix

__device__ __forceinline__ float sigmoidf_(float x) { return 1.0f / (1.0f + expf(-x)); }
__device__ __forceinline__ float siluf_(float x) { return x * sigmoidf_(x); }
__device__ __forceinline__ float softplusf_(float x) {
    return fmaxf(x, 0.0f) + log1pf(expf(-fabsf(x)));  // logaddexp(x, 0)
}

__device__ __forceinline__ v8f wmma4(v2f a, v2f b, v8f c) {
    // D = A(16x4,f32) * B(4x16,f32) + C(16x16,f32)
    return __builtin_amdgcn_wmma_f32_16x16x4_f32(false, a, false, b, (short)0, c, false, false);
}

// ---------------------------------------------------------------------------
// Pack a [Kreal x 64] row-major weight matrix into B-fragment order so that a
// lane's (b.x, b.y) pair is one contiguous float2 (-> single global_load_b64):
//   W[kk][n] -> P[ ((kk>>2)*2 + ((kk>>1)&1)) * 128 + n*2 + (kk&1) ]
// Rows kk >= Kreal are zero-filled (bakes in the K padding; no tail code).
// ---------------------------------------------------------------------------
__global__ void pack_weights(const float* __restrict__ Wsrc, float* __restrict__ P,
                             int Kreal, int Kpad) {
    int i = blockIdx.x * blockDim.x + threadIdx.x;
    if (i < Kpad * 64) {
        int kk = i >> 6, n = i & 63;
        float val = (kk < Kreal) ? Wsrc[kk * 64 + n] : 0.0f;
        int idx = ((kk >> 2) * 2 + ((kk >> 1) & 1)) * 128 + n * 2 + (kk & 1);
        P[idx] = val;
    }
}

// ---------------------------------------------------------------------------
// Node embedding GEMM: Y = node_feats @ W_emb  (bias absorbed by BN)
// Also accumulates per-column sum / sumsq into gstat[0..127].
// ---------------------------------------------------------------------------
__global__ __launch_bounds__(256) void embed_pass(
    const float* __restrict__ nf, const float* __restrict__ Wpk,
    float* __restrict__ x0, float* __restrict__ gstat, int N) {
    __shared__ float sA[TM * FV];   // 23.5 KB
    __shared__ float sStat[128];

    const int tid = threadIdx.x;
    const int r0 = blockIdx.x * TM;
    if (tid < 128) sStat[tid] = 0.0f;

    for (int i = tid; i < TM * (FV / 4); i += 256) {   // 64 rows x 23 float4
        int row = i / 23, q = i - row * 23;
        float4 val = make_float4(0.f, 0.f, 0.f, 0.f);
        if (r0 + row < N)
            val = reinterpret_cast<const float4*>(nf + (long)(r0 + row) * FV)[q];
        reinterpret_cast<float4*>(sA + row * FV)[q] = val;
    }
    __syncthreads();

    const int lane = tid & 31;
    const int ln = lane & 15;
    const int wv = tid >> 5;
    const int rs = (wv & 3) * 16;
    const int cb0 = (wv >> 2) * 32;
    const int koh = (lane < 16) ? 0 : 1;   // A/B K-pair half for this lane

    v8f acc[2];
    v8f zero = {0.f, 0.f, 0.f, 0.f, 0.f, 0.f, 0.f, 0.f};
    acc[0] = zero; acc[1] = zero;

    const float* pa = sA + (rs + ln) * FV + 2 * koh;
    const float* pb0 = Wpk + koh * 128 + (cb0 + ln) * 2;
    for (int k = 0; k < FV; k += 4) {   // K = 92 exact
        v2f a; a.x = pa[k]; a.y = pa[k + 1];
        const v2f* pb = reinterpret_cast<const v2f*>(pb0 + (k >> 1) * 128);
        #pragma unroll
        for (int t = 0; t < 2; t++) acc[t] = wmma4(a, pb[t * 16], acc[t]);
    }

    #pragma unroll
    for (int t = 0; t < 2; t++) {
        int col = cb0 + t * 16 + ln;
        float s = 0.f, q = 0.f;
        #pragma unroll
        for (int j = 0; j < 8; j++) {
            int row = rs + j + ((lane < 16) ? 0 : 8);
            float x = acc[t][j];
            s += x; q += x * x;   // padded rows are exact zeros -> no bias
            if (r0 + row < N) x0[(long)(r0 + row) * EDIM + col] = x;
        }
        atomicAdd(&sStat[col], s);
        atomicAdd(&sStat[64 + col], q);
    }
    __syncthreads();
    if (tid < 128) atomicAdd(&gstat[tid], sStat[tid]);
}

// ---------------------------------------------------------------------------
// Edge MLP GEMM.  PASS 0: stats only.  PASS 1: BN+act, gate*screen, scatter.
// hc = [v[src] | v[dst] | edge_feats] staged in LDS (64 x 172, zero padded).
// 128 threads = 4 waves; wave w owns row strip w*16..w*16+15 and computes
// BOTH matrices (4 gate tiles + 4 screen tiles, 64 acc VGPRs).  One A-frag
// LDS load feeds 8 WMMAs, and gate*screen is a pure in-register product
// (no LDS round-trip, no post-loop barriers).
// ---------------------------------------------------------------------------
template <int PASS>
__global__ __launch_bounds__(128) void edge_pass(
    const float* __restrict__ vfeat, const float* __restrict__ efeat,
    const int* __restrict__ src, const int* __restrict__ dst,
    const float* __restrict__ Wmp, const float* __restrict__ Wsp,
    const float* __restrict__ gm, const float* __restrict__ bem,
    const float* __restrict__ gs, const float* __restrict__ bes,
    float* __restrict__ gstat, const float* __restrict__ fin,
    float* __restrict__ agg, int M) {
    __shared__ float sA[TM * CP];   // 44 KB
    __shared__ float sStat[256];
    __shared__ int sIdx[2 * TM];

    const int tid = threadIdx.x;
    const int e0 = blockIdx.x * TM;

    if (PASS == 0)
        for (int i = tid; i < 256; i += 128) sStat[i] = 0.0f;
    if (tid < TM) sIdx[tid] = (e0 + tid < M) ? src[e0 + tid] : 0;
    else          sIdx[tid] = (e0 + tid - TM < M) ? dst[e0 + tid - TM] : 0;
    __syncthreads();

    // gather v[src] (cols 0..63) and v[dst] (cols 64..127), float4
    for (int i = tid; i < TM * 16 * 2; i += 128) {
        int half = (i >= TM * 16);
        int j = i & (TM * 16 - 1);
        int row = j >> 4, q = j & 15;
        float4 val = make_float4(0.f, 0.f, 0.f, 0.f);
        if (e0 + row < M)
            val = reinterpret_cast<const float4*>(vfeat + (long)sIdx[half * TM + row] * EDIM)[q];
        reinterpret_cast<float4*>(sA + row * CP + half * EDIM)[q] = val;
    }
    // edge features (cols 128..168)
    for (int i = tid; i < TM * FE; i += 128) {
        int row = i / FE, c = i - row * FE;
        sA[row * CP + 2 * EDIM + c] =
            (e0 + row < M) ? efeat[(long)(e0 + row) * FE + c] : 0.0f;
    }
    // K padding (cols 169..171)
    for (int i = tid; i < TM * 3; i += 128) {
        int row = i / 3, c = i - row * 3;
        sA[row * CP + CR + c] = 0.0f;
    }
    __syncthreads();

    const int lane = tid & 31;
    const int ln = lane & 15;
    const int wv = tid >> 5;
    const int rs = wv * 16;                 // 4 waves x 16 rows = 64
    const int koh = (lane < 16) ? 0 : 1;

    v8f am[4], as_[4];
    v8f zero = {0.f, 0.f, 0.f, 0.f, 0.f, 0.f, 0.f, 0.f};
    #pragma unroll
    for (int t = 0; t < 4; t++) { am[t] = zero; as_[t] = zero; }

    const float* pa = sA + (rs + ln) * CP + 2 * koh;
    const float* pbm0 = Wmp + koh * 128 + ln * 2;
    const float* pbs0 = Wsp + koh * 128 + ln * 2;

    for (int k = 0; k < CP; k += 4) {   // uniform 43 steps; padding baked into pack
        v2f a; a.x = pa[k]; a.y = pa[k + 1];
        const v2f* pbm = reinterpret_cast<const v2f*>(pbm0 + (k >> 1) * 128);
        const v2f* pbs = reinterpret_cast<const v2f*>(pbs0 + (k >> 1) * 128);
        #pragma unroll
        for (int t = 0; t < 4; t++) {
            am[t]  = wmma4(a, pbm[t * 16], am[t]);
            as_[t] = wmma4(a, pbs[t * 16], as_[t]);
        }
    }

    if (PASS == 0) {
        #pragma unroll
        for (int t = 0; t < 4; t++) {
            float sm = 0.f, qm = 0.f, ss = 0.f, qs = 0.f;
            #pragma unroll
            for (int j = 0; j < 8; j++) {
                float xm = am[t][j], xs = as_[t][j];
                sm += xm; qm += xm * xm;
                ss += xs; qs += xs * xs;
            }
            int col = t * 16 + ln;
            atomicAdd(&sStat[col], sm);
            atomicAdd(&sStat[64 + col], qm);
            atomicAdd(&sStat[128 + col], ss);
            atomicAdd(&sStat[192 + col], qs);
        }
        __syncthreads();
        for (int i = tid; i < 256; i += 128) atomicAdd(&gstat[i], sStat[i]);
    } else {
        #pragma unroll
        for (int t = 0; t < 4; t++) {
            int col = t * 16 + ln;
            float muM = fin[col],       rsM = fin[64 + col];
            float muS = fin[128 + col], rsS = fin[192 + col];
            float gaM = gm[col], beM = bem[col];
            float gaS = gs[col], beS = bes[col];
            #pragma unroll
            for (int j = 0; j < 8; j++) {
                int row = rs + j + ((lane < 16) ? 0 : 8);
                if (e0 + row < M) {
                    float zm = (am[t][j] - muM) * rsM * gaM + beM;
                    float zs = (as_[t][j] - muS) * rsS * gaS + beS;
                    float prod = sigmoidf_(zm) * softplusf_(zs);
                    atomicAdd(&agg[(long)sIdx[TM + row] * EDIM + col], prod);
                }
            }
        }
    }
}

// ---------------------------------------------------------------------------
__global__ void finalize_stats(const float* __restrict__ s, float* __restrict__ o,
                               int ncols, float invn) {
    int c = blockIdx.x * blockDim.x + threadIdx.x;
    if (c < ncols) {
        float m = s[c] * invn;
        float var = fmaxf(s[ncols + c] * invn - m * m, 0.0f);  // biased variance
        o[c] = m;
        o[ncols + c] = rsqrtf(var + EPS);
    }
}

__global__ void bnact_node(const float* __restrict__ x0, const float* __restrict__ fin,
                           const float* __restrict__ g, const float* __restrict__ be,
                           float* __restrict__ v, int n64) {
    int i = blockIdx.x * blockDim.x + threadIdx.x;
    if (i < n64) {
        int c = i & 63;
        float z = (x0[i] - fin[c]) * fin[64 + c] * g[c] + be[c];
        v[i] = siluf_(z);
    }
}

__global__ void colstats64(const float* __restrict__ x, float* __restrict__ gstat, int n64) {
    __shared__ float ss[128];
    if (threadIdx.x < 128) ss[threadIdx.x] = 0.0f;
    __syncthreads();
    int stride = gridDim.x * blockDim.x;
    for (int i = blockIdx.x * blockDim.x + threadIdx.x; i < n64; i += stride) {
        float val = x[i];
        int c = i & 63;
        atomicAdd(&ss[c], val);
        atomicAdd(&ss[64 + c], val * val);
    }
    __syncthreads();
    if (threadIdx.x < 128) atomicAdd(&gstat[threadIdx.x], ss[threadIdx.x]);
}

__global__ void update_node(const float* __restrict__ agg, const float* __restrict__ fin,
                            const float* __restrict__ gn, const float* __restrict__ ben,
                            float* __restrict__ v, int n64) {
    int i = blockIdx.x * blockDim.x + threadIdx.x;
    if (i < n64) {
        int c = i & 63;
        float z = (agg[i] - fin[c]) * fin[64 + c] * gn[c] + ben[c] + v[i];
        v[i] = softplusf_(z);
    }
}

__global__ void pool_kernel(const float* __restrict__ v, const int* __restrict__ gid,
                            float* __restrict__ sums, float* __restrict__ cnt, int n64) {
    int i = blockIdx.x * blockDim.x + threadIdx.x;
    if (i < n64) {
        int n = i >> 6, c = i & 63;
        int g = gid[n];
        atomicAdd(&sums[g * 64 + c], v[i]);
        if (c == 0) atomicAdd(&cnt[g], 1.0f);
    }
}

__global__ void fc0_kernel(const float* __restrict__ psum, const float* __restrict__ pcnt,
                           const float* __restrict__ Wf0, float* __restrict__ y0,
                           float* __restrict__ st) {
    __shared__ float vs[64];
    int r = blockIdx.x, c = threadIdx.x;   // 128 threads
    if (c < 64) vs[c] = psum[r * 64 + c] / fmaxf(pcnt[r], 1.0f);
    __syncthreads();
    float y = 0.0f;
    for (int k = 0; k < 64; k++) y += vs[k] * Wf0[k * 128 + c];
    y0[r * 128 + c] = y;
    atomicAdd(&st[c], y);
    atomicAdd(&st[128 + c], y * y);
}

__global__ void fc1_kernel(const float* __restrict__ y0, const float* __restrict__ fin,
                           const float* __restrict__ gf0, const float* __restrict__ bef0,
                           const float* __restrict__ Wf1, float* __restrict__ y1,
                           float* __restrict__ st) {
    __shared__ float hb[128];
    int r = blockIdx.x, c = threadIdx.x;   // 128 threads
    float yv = y0[r * 128 + c];
    hb[c] = siluf_((yv - fin[c]) * fin[128 + c] * gf0[c] + bef0[c]);
    __syncthreads();
    if (c < 64) {
        float y = 0.0f;
        for (int k = 0; k < 128; k++) y += hb[k] * Wf1[k * 64 + c];
        y1[r * 64 + c] = y;
        atomicAdd(&st[c], y);
        atomicAdd(&st[64 + c], y * y);
    }
}

__global__ void head_kernel(const float* __restrict__ y1, const float* __restrict__ fin2,
                            const float* __restrict__ gf1, const float* __restrict__ bef1,
                            const float* __restrict__ Wt, const float* __restrict__ bt,
                            float* __restrict__ out) {
    int r = threadIdx.x;   // 256 threads, 1 block
    float acc = 0.0f;
    for (int c = 0; c < 64; c++) {
        float t = siluf_((y1[r * 64 + c] - fin2[c]) * fin2[64 + c] * gf1[c] + bef1[c]);
        acc += t * Wt[c];
    }
    out[r] = acc + bt[0];
}

// ---------------------------------------------------------------------------
extern "C" void kernel_launch(void* const* d_in, const int* in_sizes, int n_in,
                              void* d_out, int out_size, void* d_ws, size_t ws_size,
                              hipStream_t stream) {
    const float* node_feats = (const float*)d_in[0];
    const float* edge_feats = (const float*)d_in[1];
    const int*   src        = (const int*)d_in[2];
    const int*   dst        = (const int*)d_in[3];
    const int*   gid        = (const int*)d_in[4];
    const float* W_emb      = (const float*)d_in[5];
    const float* g_emb      = (const float*)d_in[7];
    const float* be_emb     = (const float*)d_in[8];
    const float* Wm         = (const float*)d_in[9];
    const float* gm         = (const float*)d_in[11];
    const float* bem        = (const float*)d_in[12];
    const float* Ws         = (const float*)d_in[13];
    const float* gs         = (const float*)d_in[15];
    const float* bes        = (const float*)d_in[16];
    const float* gn         = (const float*)d_in[17];
    const float* ben        = (const float*)d_in[18];
    const float* Wf0        = (const float*)d_in[19];
    const float* gf0        = (const float*)d_in[21];
    const float* bef0       = (const float*)d_in[22];
    const float* Wf1        = (const float*)d_in[23];
    const float* gf1        = (const float*)d_in[25];
    const float* bef1       = (const float*)d_in[26];
    const float* Wt         = (const float*)d_in[27];
    const float* bt         = (const float*)d_in[28];

    const int N = in_sizes[4];          // graph_ids length
    const int M = in_sizes[2];          // src length
    const int L = in_sizes[11] / 64;    // gm is [L,64]
    const int n64 = N * EDIM;

    // workspace layout (floats)
    float* w = (float*)d_ws;
    float* vbuf  = w;                       // N*64
    float* x0agg = vbuf + (size_t)n64;      // N*64 (embedding pre-act, then per-layer agg)
    float* gstat = x0agg + (size_t)n64;     // 512
    float* fin   = gstat + 512;             // 512
    float* psum  = fin + 512;               // 256*64
    float* pcnt  = psum + NG * 64;          // 256
    float* y0    = pcnt + NG;               // 256*128
    float* y1    = y0 + NG * 128;           // 256*64
    float* pemb  = y1 + NG * 64;            // packed W_emb
    float* pedge = pemb + PK_EMB;           // packed Wm/Ws per layer: L*2*PK_EDGE

    const int gridEW = (n64 + 255) / 256;
    const int gridN  = (N + TM - 1) / TM;
    const int gridE  = (M + TM - 1) / TM;

    // --- pack all weight matrices into B-fragment order ---
    pack_weights<<<(FV * 64 + 255) / 256, 256, 0, stream>>>(W_emb, pemb, FV, FV);
    for (int l = 0; l < L; l++) {
        pack_weights<<<(CP * 64 + 255) / 256, 256, 0, stream>>>(
            Wm + (size_t)l * CR * EDIM, pedge + (size_t)(2 * l) * PK_EDGE, CR, CP);
        pack_weights<<<(CP * 64 + 255) / 256, 256, 0, stream>>>(
            Ws + (size_t)l * CR * EDIM, pedge + (size_t)(2 * l + 1) * PK_EDGE, CR, CP);
    }

    // --- node embedding: GEMM + stats, BN+silu ---
    hipMemsetAsync(gstat, 0, 512 * sizeof(float), stream);
    embed_pass<<<gridN, 256, 0, stream>>>(node_feats, pemb, x0agg, gstat, N);
    finalize_stats<<<1, 128, 0, stream>>>(gstat, fin, 64, 1.0f / (float)N);
    bnact_node<<<gridEW, 256, 0, stream>>>(x0agg, fin, g_emb, be_emb, vbuf, n64);

    // --- conv layers ---
    for (int l = 0; l < L; l++) {
        const float* Wmp = pedge + (size_t)(2 * l) * PK_EDGE;
        const float* Wsp = pedge + (size_t)(2 * l + 1) * PK_EDGE;
        const float* gml = gm + l * 64;
        const float* beml = bem + l * 64;
        const float* gsl = gs + l * 64;
        const float* besl = bes + l * 64;

        hipMemsetAsync(gstat, 0, 512 * sizeof(float), stream);
        edge_pass<0><<<gridE, 128, 0, stream>>>(vbuf, edge_feats, src, dst, Wmp, Wsp,
                                                gml, beml, gsl, besl, gstat, fin,
                                                x0agg, M);
        finalize_stats<<<1, 128, 0, stream>>>(gstat, fin, 64, 1.0f / (float)M);
        finalize_stats<<<1, 128, 0, stream>>>(gstat + 128, fin + 128, 64, 1.0f / (float)M);

        hipMemsetAsync(x0agg, 0, (size_t)n64 * sizeof(float), stream);
        edge_pass<1><<<gridE, 128, 0, stream>>>(vbuf, edge_feats, src, dst, Wmp, Wsp,
                                                gml, beml, gsl, besl, gstat, fin,
                                                x0agg, M);

        hipMemsetAsync(gstat, 0, 512 * sizeof(float), stream);
        colstats64<<<1024, 256, 0, stream>>>(x0agg, gstat, n64);
        finalize_stats<<<1, 128, 0, stream>>>(gstat, fin, 64, 1.0f / (float)N);
        update_node<<<gridEW, 256, 0, stream>>>(x0agg, fin, gn + l * 64, ben + l * 64,
                                                vbuf, n64);
    }

    // --- pooling + FC head ---
    hipMemsetAsync(psum, 0, (NG * 64 + NG) * sizeof(float), stream);
    pool_kernel<<<gridEW, 256, 0, stream>>>(vbuf, gid, psum, pcnt, n64);

    hipMemsetAsync(gstat, 0, 512 * sizeof(float), stream);
    fc0_kernel<<<NG, 128, 0, stream>>>(psum, pcnt, Wf0, y0, gstat);
    finalize_stats<<<1, 128, 0, stream>>>(gstat, fin, 128, 1.0f / (float)NG);
    fc1_kernel<<<NG, 128, 0, stream>>>(y0, fin, gf0, bef0, Wf1, y1, gstat + 256);
    finalize_stats<<<1, 128, 0, stream>>>(gstat + 256, fin + 256, 64, 1.0f / (float)NG);
    head_kernel<<<1, NG, 0, stream>>>(y1, fin + 256, gf1, bef1, Wt, bt, (float*)d_out);
}